// ResNet_62998580297732
// MI455X (gfx1250) — compile-verified
//
#include <hip/hip_runtime.h>
#include <stdint.h>

// ---------------------------------------------------------------------------
// RetinaNet post-process for MI455X (gfx1250, wave32).
//  K1 decode_kernel : decode+clip 196416 boxes (memory bound, ~6MB)
//  K2 topk_kernel   : per-class (80 blocks) histogram-select + LDS bitonic
//                     sort -> sorted top-1000 (score,idx). Dominant traffic:
//                     2 passes over 63MB of scores.
//  K3 gather_kernel : gather top boxes (AoS float4)
//  K4 mask_kernel   : per-class pairwise suppression bitmask. TDM stages the
//                     16KB box tile into LDS; V_WMMA_F32_16X16X4_F32 computes
//                     the pairwise area-sum matrix (rank-2 outer add), VALU
//                     computes intersections; iou>0.5 <=> 3*inter > areaSum.
//  K5 reduce_kernel : 1 wave/class serial greedy reduce over the bitmask
//                     (matches reference's sequential fori_loop), writes
//                     scores / labels / boxes / keep.
// ---------------------------------------------------------------------------

#define NUM_A   196416
#define NUM_C   80
#define TOPK    1000
#define KPAD    1024
#define NBINS   2048
#define CAND_CAP 2048
#define IMG_WH  1024.0f
#define CLS_THR 0.05f

typedef float    v2f __attribute__((ext_vector_type(2)));
typedef float    v8f __attribute__((ext_vector_type(8)));
typedef unsigned v4u __attribute__((ext_vector_type(4)));
typedef int      v4i __attribute__((ext_vector_type(4)));
typedef int      v8i __attribute__((ext_vector_type(8)));

#if __has_builtin(__builtin_amdgcn_wmma_f32_16x16x4_f32)
#define HAS_WMMA_F32_K4 1
#else
#define HAS_WMMA_F32_K4 0
#endif

#if __has_builtin(__builtin_amdgcn_tensor_load_to_lds)
#define HAS_TDM 1
#else
#define HAS_TDM 0
#endif

// -------------------------------- K1: decode -------------------------------
__global__ void decode_kernel(const float* __restrict__ anchors,
                              const float* __restrict__ regression,
                              float* __restrict__ boxes) {
  int a = blockIdx.x * blockDim.x + threadIdx.x;
  if (a >= NUM_A) return;
  float4 an = ((const float4*)anchors)[a];
  float4 rg = ((const float4*)regression)[a];
  float wa = an.z - an.x, ha = an.w - an.y;
  float cxa = an.x + 0.5f * wa, cya = an.y + 0.5f * ha;
  float cx = cxa + rg.x * 0.1f * wa;
  float cy = cya + rg.y * 0.1f * ha;
  float w  = expf(rg.z * 0.2f) * wa;
  float h  = expf(rg.w * 0.2f) * ha;
  float4 o;
  o.x = fminf(fmaxf(cx - 0.5f * w, 0.f), IMG_WH);
  o.y = fminf(fmaxf(cy - 0.5f * h, 0.f), IMG_WH);
  o.z = fminf(fmaxf(cx + 0.5f * w, 0.f), IMG_WH);
  o.w = fminf(fmaxf(cy + 0.5f * h, 0.f), IMG_WH);
  ((float4*)boxes)[a] = o;
}

// -------------------------------- K2: top-k --------------------------------
__device__ __forceinline__ int score_bin(float s) {
  int b = (int)(s * (float)NBINS);
  return b < 0 ? 0 : (b > NBINS - 1 ? NBINS - 1 : b);
}

__global__ void __launch_bounds__(1024) topk_kernel(const float* __restrict__ cls,
                                                    float* __restrict__ tscores,
                                                    int* __restrict__ tidx) {
  __shared__ unsigned hist[NBINS];
  __shared__ unsigned long long keys[CAND_CAP];
  __shared__ unsigned cand_count;
  __shared__ int thr_bin;
  const int c = blockIdx.x;
  const int tid = threadIdx.x;

  for (int i = tid; i < NBINS; i += 1024) hist[i] = 0u;
  for (int i = tid; i < CAND_CAP; i += 1024) keys[i] = 0ull;
  if (tid == 0) cand_count = 0u;
  __syncthreads();

  // Pass 1: value-space histogram (scores uniform in [0,1))
  for (int a = tid; a < NUM_A; a += 1024) {
    if (a + 8192 < NUM_A)
      __builtin_prefetch(cls + (size_t)(a + 8192) * NUM_C + c, 0, 0);
    float s = cls[(size_t)a * NUM_C + c];
    atomicAdd(&hist[score_bin(s)], 1u);
  }
  __syncthreads();

  if (tid == 0) {
    unsigned acc = 0; int b = NBINS - 1;
    for (; b > 0; --b) { acc += hist[b]; if (acc >= TOPK) break; }
    thr_bin = b;
  }
  __syncthreads();
  const int tb = thr_bin;

  // Pass 2: compact candidates above threshold bin
  for (int a = tid; a < NUM_A; a += 1024) {
    float s = cls[(size_t)a * NUM_C + c];
    if (score_bin(s) >= tb) {
      unsigned pos = atomicAdd(&cand_count, 1u);
      if (pos < CAND_CAP)
        keys[pos] = ((unsigned long long)__float_as_uint(s) << 32) | (unsigned)a;
    }
  }
  __syncthreads();

  // Bitonic sort, descending (positive float bits are order-preserving)
  for (unsigned k = 2; k <= CAND_CAP; k <<= 1) {
    for (unsigned j = k >> 1; j > 0; j >>= 1) {
      for (unsigned i = tid; i < CAND_CAP; i += 1024) {
        unsigned ixj = i ^ j;
        if (ixj > i) {
          unsigned long long va = keys[i], vb = keys[ixj];
          bool descBlk = ((i & k) == 0);
          bool sw = descBlk ? (va < vb) : (va > vb);
          if (sw) { keys[i] = vb; keys[ixj] = va; }
        }
      }
      __syncthreads();
    }
  }

  float s = 0.f; int idx = 0;
  if (tid < TOPK) {
    unsigned long long kv = keys[tid];
    s = __uint_as_float((unsigned)(kv >> 32));
    idx = (int)(unsigned)(kv & 0xFFFFFFFFu);
  }
  tscores[(size_t)c * KPAD + tid] = s;
  tidx[(size_t)c * KPAD + tid] = idx;
}

// -------------------------------- K3: gather -------------------------------
__global__ void gather_kernel(const float* __restrict__ boxes,
                              const float* __restrict__ tscores,
                              const int* __restrict__ tidx,
                              float* __restrict__ tboxes) {
  int t = blockIdx.x * blockDim.x + threadIdx.x;
  if (t >= NUM_C * KPAD) return;
  float s = tscores[t];
  int a = tidx[t];
  float4 b = make_float4(0.f, 0.f, 0.f, 0.f);
  if (s > 0.f) b = ((const float4*)boxes)[a];
  ((float4*)tboxes)[t] = b;
}

// ---------------------- TDM: stage 16KB class tile to LDS ------------------
__device__ __forceinline__ void tdm_load_tile(const float* gsrc, float* lds_dst) {
#if HAS_TDM
  unsigned g0[4]; int g1[8]; int gz4[4] = {0, 0, 0, 0};
  unsigned long long ga = (unsigned long long)(uintptr_t)gsrc;
  unsigned lds = (unsigned)(uintptr_t)lds_dst;   // low 32 bits = LDS byte offset
  // D# group0: count=1, lds_addr, global_addr[56:0], type=2 ("image")
  g0[0] = 1u;
  g0[1] = lds;
  g0[2] = (unsigned)ga;
  g0[3] = (unsigned)((ga >> 32) & 0x01FFFFFFu) | (2u << 30);
  // D# group1: data_size=4B; tensor_dim0=tile_dim0=4096 elems (1 row of 16KB),
  // tensor_dim1=tile_dim1=1, tensor_dim0_stride=4096
  g1[0] = (int)(2u << 16);                       // data_size = 2 (4 bytes)
  g1[1] = (int)((4096u & 0xFFFFu) << 16);        // tensor_dim0[15:0] @ bits63:48
  g1[2] = (int)((4096u >> 16) | (1u << 16));     // tensor_dim0[31:16], tensor_dim1[15:0]=1
  g1[3] = (int)(4096u << 16);                    // tensor_dim1[31:16]=0, tile_dim0=4096
  g1[4] = 1;                                     // tile_dim1=1, tile_dim2=0
  g1[5] = 4096;                                  // tensor_dim0_stride[31:0]
  g1[6] = 0; g1[7] = 0;
#if defined(__clang_major__) && (__clang_major__ >= 23)
  int gz8[8] = {0, 0, 0, 0, 0, 0, 0, 0};
  __builtin_amdgcn_tensor_load_to_lds(*(v4u*)g0, *(v8i*)g1, *(v4i*)gz4, *(v4i*)gz4,
                                      *(v8i*)gz8, 0);
#else
  __builtin_amdgcn_tensor_load_to_lds(*(v4u*)g0, *(v8i*)g1, *(v4i*)gz4, *(v4i*)gz4, 0);
#endif
#endif
}

// -------------------------- K4: suppression bitmask ------------------------
__global__ void __launch_bounds__(1024) mask_kernel(const float* __restrict__ tboxes,
                                                    unsigned* __restrict__ mask) {
  __shared__ float stage[KPAD * 4];                                  // 16 KB
  __shared__ float sx1[KPAD], sy1[KPAD], sx2[KPAD], sy2[KPAD], sar[KPAD]; // 20 KB
  const int c = blockIdx.x;
  const int tid = threadIdx.x;
  const float* src = tboxes + (size_t)c * KPAD * 4;

#if HAS_TDM
  if (tid < 32) {                 // wave 0 issues the TDM descriptor
    tdm_load_tile(src, stage);
#if __has_builtin(__builtin_amdgcn_s_wait_tensorcnt)
    __builtin_amdgcn_s_wait_tensorcnt(0);
#endif
  }
#else
  for (int i = tid; i < KPAD * 4; i += 1024) stage[i] = src[i];
#endif
  __syncthreads();

  { // AoS -> SoA + areas
    float x1 = stage[tid * 4 + 0], y1 = stage[tid * 4 + 1];
    float x2 = stage[tid * 4 + 2], y2 = stage[tid * 4 + 3];
    sx1[tid] = x1; sy1[tid] = y1; sx2[tid] = x2; sy2[tid] = y2;
    sar[tid] = (x2 - x1) * (y2 - y1);
  }
  __syncthreads();

  const int wid = tid >> 5;
  const int lane = tid & 31;
  const int lane15 = lane & 15;
  const bool lowHalf = lane < 16;

  // 2048 wave-tasks per class: 64 row-tiles(16) x 32 word-cols(32)
  for (int task = wid; task < 64 * 32; task += 32) {
    const int i0 = (task >> 5) << 4;
    const int j0 = (task & 31) << 5;
    unsigned b0[8], b1[8];

#if HAS_WMMA_F32_K4
    // A (16x4): rows [area_i, 1, 0, 0]  -> lanes0-15: {K0=area, K1=1}; lanes16-31: {K2=0,K3=0}
    v2f A;
    A.x = lowHalf ? sar[i0 + lane15] : 0.f;
    A.y = lowHalf ? 1.f : 0.f;
#endif
#pragma unroll
    for (int h = 0; h < 2; ++h) {
      const int j0h = j0 + (h << 4);
#if HAS_WMMA_F32_K4
      // B (4x16): row0 = ones, row1 = area_j  ->  D[m][n] = area_m + area_n
      v2f B;
      B.x = lowHalf ? 1.f : 0.f;
      B.y = lowHalf ? sar[j0h + lane15] : 0.f;
      v8f D = {0.f, 0.f, 0.f, 0.f, 0.f, 0.f, 0.f, 0.f};
      D = __builtin_amdgcn_wmma_f32_16x16x4_f32(false, A, false, B, (short)0, D,
                                                false, false);
#endif
      const int jj = j0h + lane15;
      const float jx1 = sx1[jj], jy1 = sy1[jj], jx2 = sx2[jj], jy2 = sy2[jj];
      const int ibase = i0 + (lowHalf ? 0 : 8);
#pragma unroll
      for (int r = 0; r < 8; ++r) {
        const int ii = ibase + r;
        const float iw = fminf(sx2[ii], jx2) - fmaxf(sx1[ii], jx1);
        const float ih = fminf(sy2[ii], jy2) - fmaxf(sy1[ii], jy1);
        const float inter = fmaxf(iw, 0.f) * fmaxf(ih, 0.f);
#if HAS_WMMA_F32_K4
        const float asum = D[r];
#else
        const float asum = sar[ii] + sar[jj];
#endif
        // iou > 0.5  <=>  3*inter > area_i + area_j (+eps); suppress only j>i
        const bool sup = (3.f * inter > asum + 1e-8f) && (jj > ii);
        const unsigned bm = __builtin_amdgcn_ballot_w32(sup);
        if (h == 0) b0[r] = bm; else b1[r] = bm;
      }
    }
    // Assemble 32-bit mask words per row (bits = col - j0)
    unsigned outw = 0;
#pragma unroll
    for (int r = 0; r < 8; ++r) {
      const unsigned wr  = (b0[r] & 0xFFFFu) | ((b1[r] & 0xFFFFu) << 16); // row i0+r
      const unsigned wr8 = (b0[r] >> 16) | (b1[r] & 0xFFFF0000u);         // row i0+r+8
      if (lane == r)     outw = wr;
      if (lane == r + 8) outw = wr8;
    }
    if (lane < 16)
      mask[(((size_t)c * KPAD) + (size_t)(i0 + lane)) * 32 + (unsigned)(task & 31)] = outw;
  }
}

// ----------------------- K5: greedy reduce + outputs -----------------------
__global__ void __launch_bounds__(32) reduce_kernel(const float* __restrict__ tscores,
                                                    const float* __restrict__ tboxes,
                                                    const unsigned* __restrict__ mask,
                                                    float* __restrict__ out) {
  __shared__ unsigned removed[32];
  const int c = blockIdx.x;
  const int lane = threadIdx.x;
  removed[lane] = 0u;
  __syncthreads();

  const float* sc = tscores + (size_t)c * KPAD;
  const unsigned* mrow = mask + (size_t)c * KPAD * 32;

  for (int i = 0; i < TOPK; ++i) {
    float si = sc[i];
    bool act = (si > CLS_THR) && !((removed[i >> 5] >> (i & 31)) & 1u);
    if (act) removed[lane] |= mrow[(size_t)i * 32 + lane];
    __syncthreads();
  }

  float* out_scores = out;
  int*   out_labels = (int*)(out + (size_t)NUM_C * TOPK);
  float* out_boxes  = out + (size_t)2 * NUM_C * TOPK;
  int*   out_keep   = (int*)(out + (size_t)6 * NUM_C * TOPK);
  const float4* tb = (const float4*)tboxes + (size_t)c * KPAD;

  for (int j = lane; j < TOPK; j += 32) {
    float s = sc[j];
    bool kept = (s > CLS_THR) && !((removed[j >> 5] >> (j & 31)) & 1u);
    int o = c * TOPK + j;
    out_scores[o] = kept ? s : 0.f;
    out_labels[o] = kept ? c : -1;
    float4 b = kept ? tb[j] : make_float4(0.f, 0.f, 0.f, 0.f);
    ((float4*)out_boxes)[o] = b;
    out_keep[o] = kept ? 1 : 0;
  }
}

// ------------------------------- launcher ----------------------------------
extern "C" void kernel_launch(void* const* d_in, const int* in_sizes, int n_in,
                              void* d_out, int out_size, void* d_ws, size_t ws_size,
                              hipStream_t stream) {
  (void)in_sizes; (void)n_in; (void)out_size; (void)ws_size;
  const float* cls = (const float*)d_in[0];
  const float* reg = (const float*)d_in[1];
  const float* anc = (const float*)d_in[2];

  char* w = (char*)d_ws;
  constexpr size_t OFF_BOXES   = 0;                                   // 196416*16
  constexpr size_t OFF_TSCORES = OFF_BOXES   + (size_t)NUM_A * 16;    // 80*1024*4
  constexpr size_t OFF_TIDX    = OFF_TSCORES + (size_t)NUM_C * KPAD * 4;
  constexpr size_t OFF_TBOXES  = OFF_TIDX    + (size_t)NUM_C * KPAD * 4;
  constexpr size_t OFF_MASK    = OFF_TBOXES  + (size_t)NUM_C * KPAD * 16;
  float*    boxes   = (float*)(w + OFF_BOXES);
  float*    tscores = (float*)(w + OFF_TSCORES);
  int*      tidx    = (int*)(w + OFF_TIDX);
  float*    tboxes  = (float*)(w + OFF_TBOXES);
  unsigned* msk     = (unsigned*)(w + OFF_MASK);

  decode_kernel<<<dim3((NUM_A + 255) / 256), dim3(256), 0, stream>>>(anc, reg, boxes);
  topk_kernel<<<dim3(NUM_C), dim3(1024), 0, stream>>>(cls, tscores, tidx);
  gather_kernel<<<dim3((NUM_C * KPAD) / 256), dim3(256), 0, stream>>>(boxes, tscores,
                                                                      tidx, tboxes);
  mask_kernel<<<dim3(NUM_C), dim3(1024), 0, stream>>>(tboxes, msk);
  reduce_kernel<<<dim3(NUM_C), dim3(32), 0, stream>>>(tscores, tboxes, msk,
                                                      (float*)d_out);
}